// QLSTM_27273042330212
// MI455X (gfx1250) — compile-verified
//
#include <hip/hip_runtime.h>
#include <math.h>

typedef __attribute__((ext_vector_type(2))) float v2f;
typedef __attribute__((ext_vector_type(8))) float v8f;

#define NQ      4
#define SVDIM   16
#define HID     128
#define FEAT    32
#define CONCAT  160
#define TILE    16
#define MAXL    8

// single-instruction v_rcp_f32 (~1 ulp) instead of IEEE div expansion
__device__ __forceinline__ float frcp(float x) {
    return __builtin_amdgcn_rcpf(x);
}
__device__ __forceinline__ float fsigmoid(float x) {
    return frcp(1.0f + __expf(-x));
}
__device__ __forceinline__ float ftanh(float x) {
    x = fminf(fmaxf(x, -15.0f), 15.0f);
    float e = __expf(2.0f * x);
    return 1.0f - 2.0f * frcp(e + 1.0f);
}

// RX gate on qubit q: new = c*old - i*s*old[idx ^ (1<<(3-q))]
__device__ __forceinline__ void rx_apply(float re[SVDIM], float im[SVDIM], int q, float theta) {
    float h = 0.5f * theta;
    float c = __cosf(h);
    float s = __sinf(h);
    int mask = 1 << (3 - q);
    float nre[SVDIM], nim[SVDIM];
#pragma unroll
    for (int i = 0; i < SVDIM; i++) {
        int j = i ^ mask;
        nre[i] = c * re[i] + s * im[j];
        nim[i] = c * im[i] - s * re[j];
    }
#pragma unroll
    for (int i = 0; i < SVDIM; i++) { re[i] = nre[i]; im[i] = nim[i]; }
}

// AngleEmbedding(RX) + NL x (RX layer + CNOT ring) + <Z_q> expvals
__device__ __forceinline__ void vqc_eval(const float y4[NQ], const float* wg, int NL, float e4[NQ]) {
    float re[SVDIM], im[SVDIM];
#pragma unroll
    for (int i = 0; i < SVDIM; i++) { re[i] = 0.0f; im[i] = 0.0f; }
    re[0] = 1.0f;
#pragma unroll
    for (int q = 0; q < NQ; q++) rx_apply(re, im, q, y4[q]);
    for (int l = 0; l < NL; l++) {
#pragma unroll
        for (int q = 0; q < NQ; q++) rx_apply(re, im, q, wg[l * NQ + q]);
        // CNOT ring: qc -> qt=(qc+1)%4 ; new[i] = old[i ^ (((i>>cb)&1)<<tb)]
#pragma unroll
        for (int qc = 0; qc < NQ; qc++) {
            int qt = (qc + 1) & 3;
            int cb = 3 - qc, tb = 3 - qt;
            float nre[SVDIM], nim[SVDIM];
#pragma unroll
            for (int i = 0; i < SVDIM; i++) {
                int j = i ^ (((i >> cb) & 1) << tb);
                nre[i] = re[j]; nim[i] = im[j];
            }
#pragma unroll
            for (int i = 0; i < SVDIM; i++) { re[i] = nre[i]; im[i] = nim[i]; }
        }
    }
#pragma unroll
    for (int q = 0; q < NQ; q++) {
        float acc = 0.0f;
#pragma unroll
        for (int i = 0; i < SVDIM; i++) {
            float p = re[i] * re[i] + im[i] * im[i];
            acc += (((i >> (3 - q)) & 1) ? -p : p);
        }
        e4[q] = acc;
    }
}

__global__ __launch_bounds__(32)
void qlstm_kernel(const float* __restrict__ x,
                  const float* __restrict__ W_in,
                  const float* __restrict__ b_in,
                  const float* __restrict__ W_out,
                  const float* __restrict__ b_out,
                  const float* __restrict__ w_f,
                  const float* __restrict__ w_i,
                  const float* __restrict__ w_u,
                  const float* __restrict__ w_o,
                  float* __restrict__ out_seq,  // [B,T,HID]
                  float* __restrict__ out_h,    // [B,HID]
                  float* __restrict__ out_c,    // [B,HID]
                  int T, int NL)
{
    __shared__ float sh_h[TILE][HID + 4];      // +4 pad: stride 132 -> conflict-free rows
    __shared__ float sh_x[TILE][FEAT];
    __shared__ float sh_y[TILE][NQ];
    __shared__ float sh_e[4][TILE][NQ];        // gate, row, q
    __shared__ float sh_Win[NQ][CONCAT];
    __shared__ float sh_bin[NQ];
    __shared__ float sh_w[4][MAXL * NQ];

    const int l     = threadIdx.x;             // 0..31, full wave
    const int lh    = l & 15;
    const int khalf = (l >> 4) * 2;            // lanes 0-15 -> K 0,1 ; lanes 16-31 -> K 2,3
    const int b0    = blockIdx.x * TILE;

    // ---- one-time staging ----
    for (int idx = l; idx < NQ * CONCAT; idx += 32)
        sh_Win[idx / CONCAT][idx % CONCAT] = W_in[idx];
    if (l < NQ) sh_bin[l] = b_in[l];
    {
        int nlw = NL * NQ;
        for (int idx = l; idx < nlw; idx += 32) {
            sh_w[0][idx] = w_f[idx];
            sh_w[1][idx] = w_i[idx];
            sh_w[2][idx] = w_u[idx];
            sh_w[3][idx] = w_o[idx];
        }
    }
    for (int idx = l; idx < TILE * (HID + 4); idx += 32)
        ((float*)sh_h)[idx] = 0.0f;

    // W_out^T chunks in WMMA B layout (loop-invariant) + bias in C layout
    v2f   bw[8];
    float cb[8];
#pragma unroll
    for (int j = 0; j < 8; j++) {
        int col = j * 16 + lh;                 // output hidden index n
        bw[j].x = W_out[col * NQ + khalf];     // B[k=khalf  ][n=col]
        bw[j].y = W_out[col * NQ + khalf + 1]; // B[k=khalf+1][n=col]
        cb[j]   = b_out[col];
    }
    // cell state in WMMA C/D layout: lane l, vgpr r -> (row = r or r+8, col = j*16+lh)
    v8f cstate[8];
#pragma unroll
    for (int j = 0; j < 8; j++)
#pragma unroll
        for (int r = 0; r < 8; r++) cstate[j][r] = 0.0f;

    const int rowbase = (l < 16) ? 0 : 8;
    const int xrow    = l >> 1;                // x staging: half-row per lane
    const int xcol0   = (l & 1) * 16;
    __syncthreads();

    for (int t = 0; t < T; t++) {
        // ---- stage x_t tile: 16 rows x 32 feats ----
        {
            const float* src = x + ((size_t)(b0 + xrow) * T + t) * FEAT + xcol0;
#pragma unroll
            for (int i = 0; i < 16; i += 4)
                *(float4*)&sh_x[xrow][xcol0 + i] = *(const float4*)(src + i);
            // warm next timestep's cacheline (global_prefetch_b8)
            if (t + 1 < T)
                __builtin_prefetch(src + FEAT, 0, 0);
        }
        __syncthreads();

        // ---- y[b][q] = pi * tanh([h,x] . W_in[q] + b_in[q]) ; 2 dot products per lane ----
        {
            int b = lh;
#pragma unroll
            for (int k2 = 0; k2 < 2; k2++) {
                int q = khalf + k2;
                float acc = sh_bin[q];
#pragma unroll 8
                for (int k = 0; k < HID; k++)  acc += sh_h[b][k] * sh_Win[q][k];
#pragma unroll 8
                for (int k = 0; k < FEAT; k++) acc += sh_x[b][k] * sh_Win[q][HID + k];
                sh_y[b][q] = 3.14159265358979f * ftanh(acc);
            }
        }
        __syncthreads();

        // ---- VQC: 4 gates x 16 rows = 64 instances, 2 per lane ----
        {
            int row = lh;
            float y4[NQ];
#pragma unroll
            for (int q = 0; q < NQ; q++) y4[q] = sh_y[row][q];
#pragma unroll
            for (int k2 = 0; k2 < 2; k2++) {
                int gate = (l >> 4) + 2 * k2;  // lanes0-15: gates 0,2 ; lanes16-31: 1,3
                float e4[NQ];
                vqc_eval(y4, &sh_w[gate][0], NL, e4);
#pragma unroll
                for (int q = 0; q < NQ; q++) sh_e[gate][row][q] = e4[q];
            }
        }
        __syncthreads();

        // ---- A matrices in WMMA A layout: lane lh = row M, VGPR/lane-half = K ----
        v2f af, ai, au, ao;
        af.x = sh_e[0][lh][khalf]; af.y = sh_e[0][lh][khalf + 1];
        ai.x = sh_e[1][lh][khalf]; ai.y = sh_e[1][lh][khalf + 1];
        au.x = sh_e[2][lh][khalf]; au.y = sh_e[2][lh][khalf + 1];
        ao.x = sh_e[3][lh][khalf]; ao.y = sh_e[3][lh][khalf + 1];

        // ---- gate GEMM via v_wmma_f32_16x16x4_f32 + LSTM elementwise ----
#pragma unroll
        for (int j = 0; j < 8; j++) {
            v8f c0;
#pragma unroll
            for (int r = 0; r < 8; r++) c0[r] = cb[j];
            v8f fD = __builtin_amdgcn_wmma_f32_16x16x4_f32(false, af, false, bw[j], (short)0, c0, false, false);
            v8f iD = __builtin_amdgcn_wmma_f32_16x16x4_f32(false, ai, false, bw[j], (short)0, c0, false, false);
            v8f uD = __builtin_amdgcn_wmma_f32_16x16x4_f32(false, au, false, bw[j], (short)0, c0, false, false);
            v8f oD = __builtin_amdgcn_wmma_f32_16x16x4_f32(false, ao, false, bw[j], (short)0, c0, false, false);
            int col = j * 16 + lh;
#pragma unroll
            for (int r = 0; r < 8; r++) {
                float ff = fsigmoid(fD[r]);
                float ii = fsigmoid(iD[r]);
                float gg = ftanh(uD[r]);
                float oo = fsigmoid(oD[r]);
                float cc = ff * cstate[j][r] + ii * gg;
                cstate[j][r] = cc;
                float hh = oo * ftanh(cc);
                int row = rowbase + r;
                sh_h[row][col] = hh;
                out_seq[((size_t)(b0 + row) * T + t) * HID + col] = hh;
            }
        }
        // next-iter barrier (after x staging) orders sh_h writes vs. reads
    }

    __syncthreads();
    // final h (from LDS)
    for (int idx = l; idx < TILE * HID; idx += 32) {
        int row = idx >> 7, col = idx & 127;
        out_h[(size_t)(b0 + row) * HID + col] = sh_h[row][col];
    }
    // final c (from registers, C/D layout)
#pragma unroll
    for (int j = 0; j < 8; j++) {
        int col = j * 16 + lh;
#pragma unroll
        for (int r = 0; r < 8; r++)
            out_c[(size_t)(b0 + rowbase + r) * HID + col] = cstate[j][r];
    }
}

extern "C" void kernel_launch(void* const* d_in, const int* in_sizes, int n_in,
                              void* d_out, int out_size, void* d_ws, size_t ws_size,
                              hipStream_t stream) {
    const float* x     = (const float*)d_in[0];
    const float* W_in  = (const float*)d_in[1];
    const float* b_in  = (const float*)d_in[2];
    const float* W_out = (const float*)d_in[3];
    const float* b_out = (const float*)d_in[4];
    const float* w_f   = (const float*)d_in[5];
    const float* w_i   = (const float*)d_in[6];
    const float* w_u   = (const float*)d_in[7];
    const float* w_o   = (const float*)d_in[8];

    // Derive shapes: H from W_out (H*NQ), F from W_in (NQ*(F+H)), then B,T.
    const int H  = in_sizes[3] / NQ;                 // 128
    const int F  = in_sizes[1] / NQ - H;             // 32
    const long long BT = (long long)in_sizes[0] / F; // B*T
    const long long B  = ((long long)out_size - BT * H) / (2LL * H);
    const int T  = (int)(BT / B);
    int NL = in_sizes[5] / NQ;
    if (NL > MAXL) NL = MAXL;

    float* out_seq = (float*)d_out;
    float* out_h   = out_seq + (size_t)B * T * H;
    float* out_c   = out_h + (size_t)B * H;

    const int nblocks = (int)(B / TILE);             // 1024/16 = 64 waves, fully independent
    qlstm_kernel<<<nblocks, 32, 0, stream>>>(x, W_in, b_in, W_out, b_out,
                                             w_f, w_i, w_u, w_o,
                                             out_seq, out_h, out_c, T, NL);
}